// EquivariantTransposeConv_51926154609218
// MI455X (gfx1250) — compile-verified
//
#include <hip/hip_runtime.h>

// ---------------------------------------------------------------------------
// EquivariantTransposeConv for MI455X (gfx1250, wave32, WMMA)
//
// Design:
//  * Fused: never materialize feat_hr / ctx (would be 2x92MB roundtrips).
//    Both are <=5x5 low-res-tap separable filters per hi-res pixel (phase
//    tables precomputed from tconv_weight / spatial_weights, which are
//    separable for the given inputs).
//  * Tensor product out[k] = sum_ij T[i,j,k] x_i y_j recast as GEMM over the
//    outer product P[(i,j)] = x_i*y_j:  (16 pixels x 484) @ (484 x 32pad)
//    using V_WMMA_F32_16X16X4_F32, K = 484 = 121 chunks of 4.
//  * K-ordering trick: k = i*22 + j. Lanes 0-15 (A-matrix K=0,1) own
//    i in [0,11); lanes 16-31 (K=2,3) own i in [11,22). A operands are
//    produced by 1 v_pk_mul per chunk, no cross-lane data movement.
//  * W3J (not a kernel input) computed on-device in double precision
//    (CG coeffs + real<->complex basis change), once per launch, parallel.
//  * B matrix (484x32, WMMA-layout, 62KB) built once in workspace; read via
//    coalesced b64 loads (L2/WGP$ resident), prefetched with
//    __builtin_prefetch -> global_prefetch_b8.
//  * Round-1 fix: opaque loop-variant B offset defeats LICM, which previously
//    hoisted all 242 B loads (484 VGPRs) and spilled them to scratch.
//  * Round-2 fix: the opaque value is now a scalar *offset*, not the pointer,
//    so address-space inference keeps the loads on the GLOBAL path
//    (global_load_b64, LOADcnt only) instead of FLAT (LOADcnt+DScnt).
//    Output stores are non-temporal (write-once stream).
// ---------------------------------------------------------------------------

typedef __attribute__((ext_vector_type(2))) float v2f;
typedef __attribute__((ext_vector_type(4))) float v4f;
typedef __attribute__((ext_vector_type(8))) float v8f;

#define Hc  128
#define Wc  128
#define HR  512
#define WR  512
#define CF  22

// workspace layout (bytes)
#define WS_B_OFF     0           // float Bbuf[121][2][32][2] = 61952 B
#define WS_TBL_OFF   65536       // float tbl[64]: [0..29] ctx variants, [32..51] x-phase
#define WS_W3J_OFF   131072      // double, 8 paths * 2304 stride
#define WS_W3J_STR   2304        // doubles per path
#define WS_SCR_OFF   294912     // per-path scratch, 11264 doubles stride
#define WS_SCR_STR   11264

__device__ __forceinline__ double dfact(int n){
  double r = 1.0;
  for(int i = 2; i <= n; ++i) r *= (double)i;
  return r;
}

__device__ double cgcoeff(int j1,int m1,int j2,int m2,int j3,int m3){
  if(m3 != m1 + m2) return 0.0;
  int vmin = -j1 + j2 + m3; if(-j1 + m1 > vmin) vmin = -j1 + m1; if(0 > vmin) vmin = 0;
  int vmax = j2 + j3 + m1; if(j3 - j1 + j2 < vmax) vmax = j3 - j1 + j2; if(j3 + m3 < vmax) vmax = j3 + m3;
  double c = sqrt((2.0*j3 + 1.0) *
      dfact(j3+j1-j2)*dfact(j3-j1+j2)*dfact(j1+j2-j3)*dfact(j3+m3)*dfact(j3-m3) /
      (dfact(j1+j2+j3+1)*dfact(j1-m1)*dfact(j1+m1)*dfact(j2-m2)*dfact(j2+m2)));
  double s = 0.0;
  for(int v = vmin; v <= vmax; ++v){
    double sg = ((v + j2 + m2) & 1) ? -1.0 : 1.0;
    s += sg * dfact(j2+j3+m1-v)*dfact(j1-m1+v) /
         (dfact(v)*dfact(j3-j1+j2-v)*dfact(j3+m3-v)*dfact(v+j1-j2-m3));
  }
  return c * s;
}

// real->complex change of basis, times (-i)^l (l even -> real factor)
__device__ void fill_q(int l, double* qre, double* qim){
  int d = 2*l + 1;
  const double is2 = 0.70710678118654752440;
  for(int i = 0; i < d*d; ++i){ qre[i] = 0.0; qim[i] = 0.0; }
  for(int m = -l; m < 0; ++m){
    qre[(l+m)*d + (l-m)] = is2;
    qim[(l+m)*d + (l+m)] = -is2;
  }
  qre[l*d + l] = 1.0;
  for(int m = 1; m <= l; ++m){
    double sg = (m & 1) ? -1.0 : 1.0;
    qre[(l+m)*d + (l+m)] = sg * is2;
    qim[(l+m)*d + (l-m)] = sg * is2;
  }
  double f = ((l/2) & 1) ? -1.0 : 1.0;   // (-i)^l for even l
  for(int i = 0; i < d*d; ++i){ qre[i] *= f; qim[i] *= f; }
}

// one block per path: compute normalized real Wigner-3j tensor (double)
__global__ __launch_bounds__(256) void w3j_setup_kernel(char* ws){
  int path = blockIdx.x, tid = threadIdx.x;
  int l1 = (path & 4) ? 6 : 4, l2 = (path & 2) ? 6 : 4, l3 = (path & 1) ? 6 : 4;
  int d1 = 2*l1+1, d2 = 2*l2+1, d3 = 2*l3+1, n = d1*d2*d3;
  __shared__ double q1re[169], q1im[169], q2re[169], q2im[169], q3re[169], q3im[169];
  __shared__ double nscale;
  double* scr = (double*)(ws + WS_SCR_OFF) + (size_t)path * WS_SCR_STR;
  double* C  = scr;          // n real
  double* S1 = scr + 2208;   // n complex
  double* S2 = scr + 6608;   // n complex
  double* W  = (double*)(ws + WS_W3J_OFF) + (size_t)path * WS_W3J_STR;
  if(tid == 0){ fill_q(l1,q1re,q1im); fill_q(l2,q2re,q2im); fill_q(l3,q3re,q3im); }
  __syncthreads();
  for(int idx = tid; idx < n; idx += 256){
    int c = idx % d3, bq = (idx/d3) % d2, aq = idx/(d3*d2);
    C[idx] = cgcoeff(l1, aq - l1, l2, bq - l2, l3, c - l3);
  }
  __syncthreads();
  // S1[j,b,c] = sum_a Q1[a,j] * C[a,b,c]
  for(int idx = tid; idx < n; idx += 256){
    int c = idx % d3, bq = (idx/d3) % d2, j = idx/(d3*d2);
    double re = 0.0, im = 0.0;
    for(int aq = 0; aq < d1; ++aq){
      double cv = C[(aq*d2 + bq)*d3 + c];
      re += q1re[aq*d1 + j]*cv; im += q1im[aq*d1 + j]*cv;
    }
    S1[2*idx] = re; S1[2*idx+1] = im;
  }
  __syncthreads();
  // S2[j,l,c] = sum_b Q2[b,l] * S1[j,b,c]
  for(int idx = tid; idx < n; idx += 256){
    int c = idx % d3, l = (idx/d3) % d2, j = idx/(d3*d2);
    double re = 0.0, im = 0.0;
    for(int bq = 0; bq < d2; ++bq){
      int s = 2*((j*d2 + bq)*d3 + c);
      double ar = S1[s], ai = S1[s+1];
      double br = q2re[bq*d2 + l], bi = q2im[bq*d2 + l];
      re += br*ar - bi*ai; im += br*ai + bi*ar;
    }
    S2[2*idx] = re; S2[2*idx+1] = im;
  }
  __syncthreads();
  // W[j,l,m] = Re( sum_c conj(Q3[c,m]) * S2[j,l,c] )
  for(int idx = tid; idx < n; idx += 256){
    int m = idx % d3, l = (idx/d3) % d2, j = idx/(d3*d2);
    double re = 0.0;
    for(int c = 0; c < d3; ++c){
      int s = 2*((j*d2 + l)*d3 + c);
      re += q3re[c*d3 + m]*S2[s] + q3im[c*d3 + m]*S2[s+1];
    }
    W[idx] = re;
  }
  __syncthreads();
  if(tid == 0){
    double ss = 0.0;
    for(int i = 0; i < n; ++i) ss += W[i]*W[i];
    nscale = 1.0 / sqrt(ss);
  }
  __syncthreads();
  for(int idx = tid; idx < n; idx += 256) W[idx] *= nscale;
}

// Build per-phase filter tables + WMMA-layout B matrix (484 x 32pad).
__global__ __launch_bounds__(256) void bbuf_setup_kernel(char* ws, const float* __restrict__ tconv,
                                                         const float* __restrict__ spatial,
                                                         const float* __restrict__ tpw){
  float* Bb  = (float*)(ws + WS_B_OFF);
  float* tbl = (float*)(ws + WS_TBL_OFF);
  int tid = threadIdx.x;
  if(tid == 0){
    // separable 1D factors (inputs are separable: outer(b1,b1), uniform 3x3)
    double w1[6], s1[3];
    double dref = sqrt((double)tconv[2*6 + 2]);
    for(int k = 0; k < 6; ++k) w1[k] = (double)tconv[k*6 + 2] / dref;
    double sref = sqrt((double)spatial[1*3 + 1]);
    for(int d = 0; d < 3; ++d) s1[d] = (double)spatial[d*3 + 1] / sref;
    // ctx 5-tap variants: 0..3 interior phase, 4 = edge o==0, 5 = edge o==HR-1
    for(int v = 0; v < 6; ++v){
      int o = (v < 4) ? (8 + v) : ((v == 4) ? 0 : HR - 1);
      int m0 = o >> 2;
      double acc[5] = {0,0,0,0,0};
      for(int dl = -1; dl <= 1; ++dl){
        int oc = o + dl; oc = oc < 0 ? 0 : (oc > HR-1 ? HR-1 : oc);
        int mc = oc >> 2, pc = oc & 3;
        for(int d = -1; d <= 1; ++d){
          int k = 4*d + 4 - pc;                 // corr with flipped kernel
          if(k >= 0 && k <= 5) acc[mc + d - (m0 - 2)] += s1[dl+1] * w1[5 - k];
        }
      }
      for(int s = 0; s < 5; ++s) tbl[v*5 + s] = (float)acc[s];
    }
    // feat_hr 3-tap (stored 5-wide, aligned with ctx taps)
    for(int p = 0; p < 4; ++p){
      float xa[5] = {0,0,0,0,0};
      for(int d = -1; d <= 1; ++d){
        int k = 4*d + 4 - p;
        if(k >= 0 && k <= 5) xa[d + 2] = (float)w1[5 - k];
      }
      for(int s = 0; s < 5; ++s) tbl[32 + p*5 + s] = xa[s];
    }
    for(int s = 52; s < 64; ++s) tbl[s] = 0.f;
  }
  // B matrix, WMMA 4x16 f32 layout: Bbuf[t][nt][lane][v]
  //  lane<16:  rows K=0(v0),K=1(v1), N=lane ; lane>=16: rows K=2,K=3, N=lane-16
  const double PW4v = 1.5, PW6v = 1.80277563773199464656;  // sqrt(9/4), sqrt(13/4)
  for(int idx = tid; idx < 121*2*32*2; idx += 256){
    int v = idx & 1, lane = (idx >> 1) & 31, nt = (idx >> 6) & 1, t = idx >> 7;
    int e00 = (t/11)*22 + 2*(t%11);
    int e = e00 + ((lane >= 16) ? 242 : 0) + v;     // k = i*22 + j
    int nn = nt*16 + (lane & 15);
    float val = 0.f;
    if(nn < CF){
      int i = e / 22, j = e % 22, k = nn;
      int i4 = (i < 9), j4 = (j < 9), k4 = (k < 9);
      int path = (i4 ? 0 : 4) + (j4 ? 0 : 2) + (k4 ? 0 : 1);
      int d2 = j4 ? 9 : 13, d3 = k4 ? 9 : 13;
      int ii = i4 ? i : i - 9, jj = j4 ? j : j - 9, kk = k4 ? k : k - 9;
      const double* W = (const double*)(ws + WS_W3J_OFF) + (size_t)path * WS_W3J_STR;
      val = (float)((double)tpw[path] * (k4 ? PW4v : PW6v) * W[(ii*d2 + jj)*d3 + kk]);
    }
    Bb[idx] = val;
  }
}

// Fused upsample + context + tensor-product (WMMA) kernel.
// Block = 256 threads (8 waves), tile = 4 hi-res rows x 256 cols.
__global__ __launch_bounds__(256) void eqtc_main_kernel(const float* __restrict__ f4,
                                                        const float* __restrict__ f6,
                                                        const char* __restrict__ ws,
                                                        float* __restrict__ out, int Bn){
  __shared__ __align__(16) float stage[5][68][24];  // low-res rows a-2..a+2, cols m-2.., 22ch pad24
  __shared__ float tbl[64];
  const float* Bb = (const float*)(ws + WS_B_OFF);
  int tid = threadIdx.x, bz = blockIdx.x;
  int ct = bz & 1, a = (bz >> 1) & 127, b = bz >> 8;
  if(tid < 64) tbl[tid] = ((const float*)(ws + WS_TBL_OFF))[tid];
  // warm B (62KB, WGP$/L2 resident) with CDNA5 prefetch
  for(int off = tid * 128; off < 121*2*32*2*4; off += 256*128)
    __builtin_prefetch((const char*)Bb + off, 0, 0);
  // stage input tile
  for(int idx = tid; idx < 5*68*24; idx += 256){
    int ch = idx % 24, cx = (idx/24) % 68, ry = idx/(24*68);
    int iy = a - 2 + ry, ix = ct*64 - 2 + cx;
    float v = 0.f;
    if(ch < CF && (unsigned)iy < (unsigned)Hc && (unsigned)ix < (unsigned)Wc){
      int pos = b*Hc*Wc + iy*Wc + ix;
      v = (ch < 9) ? f4[pos*9 + ch] : f6[pos*13 + (ch - 9)];
    }
    stage[ry][cx][ch] = v;
  }
  __syncthreads();

  int lane = tid & 31, wave = tid >> 5, half = lane >> 4, lm = lane & 15;
  float* o6 = out + (size_t)Bn * (HR*WR) * 9;

  #pragma unroll 1
  for(int q = 0; q < 8; ++q){
    int idx = q*128 + wave*16;
    int row_in = idx >> 8, col0 = idx & 255;
    int oy = a*4 + row_in;
    int colbase = ct*256 + col0;
    int ox = colbase + lm;                 // pixel owned by this half-lane
    int px = ox & 3, py = oy & 3;
    int vx = (ox == 0) ? 4 : ((ox == WR-1) ? 5 : px);
    int vy = (oy == 0) ? 4 : ((oy == HR-1) ? 5 : py);
    float CWx[5], CWy[5], XWx[5], XWy[5];
    #pragma unroll
    for(int s = 0; s < 5; ++s){
      CWx[s] = tbl[vx*5 + s]; CWy[s] = tbl[vy*5 + s];
      XWx[s] = tbl[32 + px*5 + s]; XWy[s] = tbl[32 + py*5 + s];
    }
    int c0 = (ox >> 2) - ct*64;           // stage col of lowres m-2

    // x = feat_hr, y = ctx : separable 5x5 low-res gathers, 4ch at a time
    float x22[24], y22[24];
    #pragma unroll
    for(int cb = 0; cb < 24; cb += 4){
      v4f xs = {0,0,0,0}, ys = {0,0,0,0};
      #pragma unroll
      for(int ry = 0; ry < 5; ++ry){
        v4f rx = {0,0,0,0}, rc = {0,0,0,0};
        #pragma unroll
        for(int cx = 0; cx < 5; ++cx){
          v4f v = *(const v4f*)&stage[ry][c0 + cx][cb];
          rx += XWx[cx] * v; rc += CWx[cx] * v;
        }
        xs += XWy[ry] * rx; ys += CWy[ry] * rc;
      }
      x22[cb+0]=xs.x; x22[cb+1]=xs.y; x22[cb+2]=xs.z; x22[cb+3]=xs.w;
      y22[cb+0]=ys.x; y22[cb+1]=ys.y; y22[cb+2]=ys.z; y22[cb+3]=ys.w;
    }
    // half0 owns i in [0,11); half1 owns i in [11,22)
    float xh[11];
    #pragma unroll
    for(int s = 0; s < 11; ++s) xh[s] = half ? x22[s + 11] : x22[s];

    // Opaque, loop-variant scalar OFFSET (not pointer!) into B:
    //  * defeats LICM (no hoist of 242 loads -> no 484-VGPR live range/spills)
    //  * keeps the base pointer traceable to the kernarg, so the loads lower
    //    to global_load_b64 (LOADcnt only), not flat_load_b64 (LOADcnt+DScnt)
    int qoff = 0;
    asm volatile("" : "+s"(qoff));
    const float* Bq = Bb + qoff;

    // GEMM: (16px x 484) @ (484 x 32), K chunks of 4, fp32 WMMA
    v8f acc0 = {}, acc1 = {};
    #pragma unroll
    for(int il = 0; il < 11; ++il){
      float xi = xh[il];
      #pragma unroll
      for(int jc = 0; jc < 11; ++jc){
        int t = il*11 + jc;
        v2f A; A.x = xi * y22[2*jc]; A.y = xi * y22[2*jc + 1];
        v2f B0 = *(const v2f*)(Bq + t*128 + lane*2);
        v2f B1 = *(const v2f*)(Bq + t*128 + 64 + lane*2);
        acc0 = __builtin_amdgcn_wmma_f32_16x16x4_f32(false, A, false, B0, (short)0, acc0, false, false);
        acc1 = __builtin_amdgcn_wmma_f32_16x16x4_f32(false, A, false, B1, (short)0, acc1, false, false);
      }
    }
    // D layout: lane -> N (=channel), VGPR g -> M (=pixel), M = 8*half + g
    // Output is a write-once stream: non-temporal stores keep B/input cached.
    int ch0 = lm;
    #pragma unroll
    for(int g = 0; g < 8; ++g){
      int colM = colbase + half*8 + g;
      size_t pix = (size_t)b*(HR*WR) + (size_t)oy*WR + colM;
      float v0 = acc0[g];
      if(ch0 < 9) __builtin_nontemporal_store(v0, &out[pix*9 + ch0]);
      else        __builtin_nontemporal_store(v0, &o6[pix*13 + (ch0 - 9)]);
      if(lm < 6)  __builtin_nontemporal_store(acc1[g], &o6[pix*13 + (7 + lm)]);
    }
  }
}

extern "C" void kernel_launch(void* const* d_in, const int* in_sizes, int n_in,
                              void* d_out, int out_size, void* d_ws, size_t ws_size,
                              hipStream_t stream){
  const float* f4      = (const float*)d_in[0];
  const float* f6      = (const float*)d_in[1];
  const float* tconv   = (const float*)d_in[2];
  const float* spatial = (const float*)d_in[3];
  const float* tpw     = (const float*)d_in[4];
  (void)n_in; (void)out_size; (void)ws_size;
  int Bn = in_sizes[0] / (9 * Hc * Wc);
  char* ws = (char*)d_ws;
  hipLaunchKernelGGL(w3j_setup_kernel, dim3(8), dim3(256), 0, stream, ws);
  hipLaunchKernelGGL(bbuf_setup_kernel, dim3(1), dim3(256), 0, stream, ws, tconv, spatial, tpw);
  hipLaunchKernelGGL(eqtc_main_kernel, dim3(Bn * 256), dim3(256), 0, stream, f4, f6, ws, (float*)d_out, Bn);
}